// ConsistencyLoss_73237782331455
// MI455X (gfx1250) — compile-verified
//
#include <hip/hip_runtime.h>
#include <stdint.h>

// ConsistencyLoss (KL vs class-conditional target table) for MI455X (gfx1250).
//
// Roofline: only emotion_logits (B*7*4 = 112MB) + targets (16MB) are live
// (fatigue_logits is dead in the reference). 128MB / 23.3 TB/s ~= 5.5us floor;
// ~14 transcendentals/row overlap under that. => streaming, memory-bound.
// CDNA5 path: GLOBAL_LOAD_ASYNC_TO_LDS_B128 (ASYNCcnt) double-buffered tile
// staging — the 28B row stride defeats aligned per-lane vector loads, so we
// stage perfectly-contiguous 28672B tiles into LDS and read rows from there
// (dword index 7t+j is bijective mod 64 -> bank-conflict-free).

#define TPB     256
#define TROWS   1024                 // rows per tile
#define TILE_BYTES (TROWS * 28)      // 28672 = 7 * (TPB*16)
#define NCHUNK  7                    // async B128 chunks per thread per tile

__device__ __constant__ float RAWTAB[4][7] = {
  {0.05f, 0.02f, 0.03f, 0.4f,  0.05f, 0.4f,  0.05f},
  {0.05f, 0.05f, 0.05f, 0.05f, 0.3f,  0.05f, 0.45f},
  {0.1f,  0.15f, 0.2f,  0.02f, 0.35f, 0.03f, 0.15f},
  {1.0f/7.0f, 1.0f/7.0f, 1.0f/7.0f, 1.0f/7.0f, 1.0f/7.0f, 1.0f/7.0f, 1.0f/7.0f},
};

// Async global -> LDS, 16B per lane. VDST = per-lane LDS byte address,
// VADDR = per-lane 64-bit global address (GV mode, saddr = off).
__device__ __forceinline__ void async_load16(uint32_t lds_off, uint64_t gaddr) {
  asm volatile("global_load_async_to_lds_b128 %0, %1, off"
               :: "v"(lds_off), "v"(gaddr) : "memory");
}

__device__ __forceinline__ void wait_async_le7() {
#if __has_builtin(__builtin_amdgcn_s_wait_asynccnt)
  __builtin_amdgcn_s_wait_asynccnt(7);
#else
  asm volatile("s_wait_asynccnt 0x7" ::: "memory");
#endif
}

__device__ __forceinline__ void wait_async_le0() {
#if __has_builtin(__builtin_amdgcn_s_wait_asynccnt)
  __builtin_amdgcn_s_wait_asynccnt(0);
#else
  asm volatile("s_wait_asynccnt 0x0" ::: "memory");
#endif
}

__global__ __launch_bounds__(TPB) void kl_main_kernel(
    const float* __restrict__ emo,   // [B,7]
    const int*   __restrict__ tgt,   // [B]
    float*       __restrict__ partial,
    int B, int ntiles) {
  __shared__ float smem[2][TROWS * 7];   // 2 x 28672 B staging buffers
  __shared__ float wtab[4][8];           // [r][0..6]=norm weights, [r][7]=sum w*log w
  __shared__ float red[TPB];

  const int tid = threadIdx.x;

  // Build the eps-normalized target table once per block (matches reference f32 math).
  if (tid < 4) {
    float t[7]; float s = 0.0f;
    #pragma unroll
    for (int j = 0; j < 7; ++j) { t[j] = RAWTAB[tid][j] + 1e-8f; s += t[j]; }
    float inv = 1.0f / s;
    float c = 0.0f;
    #pragma unroll
    for (int j = 0; j < 7; ++j) {
      float w = t[j] * inv;
      wtab[tid][j] = w;
      c += w * logf(w);                 // precise log, 28 calls per block total
    }
    wtab[tid][7] = c;
  }
  // (published by the first __syncthreads below)

  const uint32_t lds0 = (uint32_t)(uintptr_t)&smem[0][0];  // low 32 bits = LDS byte addr
  const uint32_t lds1 = (uint32_t)(uintptr_t)&smem[1][0];
  const uint64_t ebase  = (uint64_t)(uintptr_t)emo;
  const uint64_t maxOff = (uint64_t)B * 28u - 16u;          // last in-bounds 16B chunk

  auto issue_tile = [&](int tile, uint32_t ldsbuf) {
    uint64_t tb = (uint64_t)tile * (uint64_t)TILE_BYTES;
    #pragma unroll
    for (int c = 0; c < NCHUNK; ++c) {
      uint32_t lo  = (uint32_t)(c * (TPB * 16) + tid * 16);
      uint64_t off = tb + (uint64_t)lo;
      if (off > maxOff) off = maxOff;    // clamp tail tile in-bounds (data unused)
      async_load16(ldsbuf + lo, ebase + off);
    }
  };

  float acc = 0.0f;
  int tile = blockIdx.x;
  const int stride = (int)gridDim.x;
  if (tile < ntiles) issue_tile(tile, lds0);
  int bufsel = 0;

  for (; tile < ntiles; tile += stride) {
    const int nxt = tile + stride;
    const bool has_next = (nxt < ntiles);
    if (has_next) {
      issue_tile(nxt, bufsel ? lds0 : lds1);
      wait_async_le7();                  // the 7 loads of the CURRENT tile retired
    } else {
      wait_async_le0();
    }
    __syncthreads();                     // current tile visible block-wide (+ wtab on iter 0)

    const float* buf = bufsel ? &smem[1][0] : &smem[0][0];
    const int tstart = tile * TROWS;
    #pragma unroll
    for (int c = 0; c < 4; ++c) {
      const int lrow = tid + TPB * c;    // lanes read dword index 7t+j: bijective mod 64
      const int grow = tstart + lrow;
      if (grow < B) {
        int tg = __builtin_nontemporal_load(tgt + grow);  // single-use: th:TH_LOAD_NT
        int r  = ((unsigned)tg <= 2u) ? tg : 3;

        float x[7];
        #pragma unroll
        for (int j = 0; j < 7; ++j) x[j] = buf[lrow * 7 + j];

        float m = fmaxf(fmaxf(fmaxf(x[0], x[1]), fmaxf(x[2], x[3])),
                        fmaxf(fmaxf(x[4], x[5]), x[6]));
        float e[7]; float S = 0.0f;
        #pragma unroll
        for (int j = 0; j < 7; ++j) { e[j] = __expf(x[j] - m); S += e[j]; }
        float rs = __builtin_amdgcn_rcpf(S);

        const float* w = wtab[r];
        float dot = 0.0f;
        #pragma unroll
        for (int j = 0; j < 7; ++j)
          dot = fmaf(w[j], __logf(fmaf(e[j], rs, 1e-8f)), dot);
        acc += w[7] - dot;               // sum_j t*(log t - log(p+eps))
      }
    }
    __syncthreads();                     // all reads done before buffer is re-issued
    bufsel ^= 1;
  }

  // Block tree-reduction -> deterministic per-block partial.
  red[tid] = acc;
  __syncthreads();
  #pragma unroll
  for (int s = TPB / 2; s > 0; s >>= 1) {
    if (tid < s) red[tid] += red[tid + s];
    __syncthreads();
  }
  if (tid == 0) partial[blockIdx.x] = red[0];
}

__global__ __launch_bounds__(TPB) void kl_finalize_kernel(
    const float* __restrict__ partial, int n,
    float* __restrict__ out, float invB) {
  __shared__ float red[TPB];
  const int tid = threadIdx.x;
  float a = 0.0f;
  for (int i = tid; i < n; i += TPB) a += partial[i];
  red[tid] = a;
  __syncthreads();
  #pragma unroll
  for (int s = TPB / 2; s > 0; s >>= 1) {
    if (tid < s) red[tid] += red[tid + s];
    __syncthreads();
  }
  if (tid == 0) out[0] = red[0] * invB;
}

extern "C" void kernel_launch(void* const* d_in, const int* in_sizes, int n_in,
                              void* d_out, int out_size, void* d_ws, size_t ws_size,
                              hipStream_t stream) {
  // d_in[0] = fatigue_logits (UNUSED by the reference — never touched: saves 48MB traffic)
  const float* emo = (const float*)d_in[1];
  const int*   tgt = (const int*)d_in[2];
  float*       out = (float*)d_out;
  float*       partial = (float*)d_ws;

  const int B = in_sizes[1] / 7;
  const int ntiles = (B + TROWS - 1) / TROWS;

  int nblocks = 1024;
  if (nblocks > ntiles) nblocks = ntiles;
  const int maxb = (int)(ws_size / sizeof(float));
  if (nblocks > maxb) nblocks = maxb;
  if (nblocks < 1) nblocks = 1;

  kl_main_kernel<<<nblocks, TPB, 0, stream>>>(emo, tgt, partial, B, ntiles);
  kl_finalize_kernel<<<1, TPB, 0, stream>>>(partial, nblocks, out, 1.0f / (float)B);
}